// SDPLayer_20315195310091
// MI455X (gfx1250) — compile-verified
//
#include <hip/hip_runtime.h>
#include <math.h>

// ---------------------------------------------------------------------------
// ADMM SDP solver, eigh replaced by Newton-Schulz matrix-sign (all GEMMs).
//   proj_psd(M) = 0.5*(M + sign(M)*M),  sign via Y <- 1.5*Y - 0.5*Y*(Y*Y)
// GEMMs run on CDNA5 V_WMMA_F32_16X16X4_F32 (full-fp32 matrix path).
// 64x128 block tile, 4 waves, double-buffered LDS software pipeline.
// ---------------------------------------------------------------------------

#define NMAT 1024
#define BM 64
#define BN 128
#define BK 32
#define APAD 4
#define BPAD 4
#define NS_ITERS 16
#define ADMM_ITERS 80

typedef __attribute__((ext_vector_type(2))) float v2f;
typedef __attribute__((ext_vector_type(8))) float v8f;

// ---------------------------------------------------------------------------
// GEMM: D = A * B   (1024x1024x1024, row-major fp32)
//  transformB != 0 : B'[k][n] = 1.5*(k==n) - 0.5*B[k][n]   (Newton-Schulz)
//  epimode    == 1 : D = 0.5*(A*B + E)                      (sign apply)
// 128 threads = 4 waves; each wave: 64 rows x 32 cols = 4x2 WMMA 16x16 tiles.
// Grid: (1024/BN, 1024/BM) = (8,16) = 128 blocks.
// ---------------------------------------------------------------------------
__global__ __launch_bounds__(128)
void gemm_wmma_f32(const float* __restrict__ A, const float* __restrict__ B,
                   float* __restrict__ D, const float* __restrict__ E,
                   int transformB, int epimode)
{
    __shared__ float As[2][BM][BK + APAD];   // 2 x 9 KB
    __shared__ float Bs[2][BK][BN + BPAD];   // 2 x 16.5 KB

    const int tid  = threadIdx.x;
    const int lane = tid & 31;
    const int wc   = tid >> 5;         // wave = 32-col slab, 0..3
    const int half = lane >> 4;        // K-pair / M+8 selector
    const int l16  = lane & 15;

    const int brow = blockIdx.y * BM;
    const int bcol = blockIdx.x * BN;

    v8f acc[4][2] = {};
    float4 aR[4], bR[8];               // global->LDS staging registers

    // ---- global fetch of one K-stage into registers -----------------------
    auto loadStage = [&](int kb) {
#pragma unroll
        for (int i = 0; i < 4; ++i) {                 // A: 64x32
            int idx = tid + i * 128;
            int r = idx >> 3, c4 = (idx & 7) * 4;
            aR[i] = *(const float4*)(A + (size_t)(brow + r) * NMAT + kb + c4);
        }
#pragma unroll
        for (int i = 0; i < 8; ++i) {                 // B: 32x128
            int idx = tid + i * 128;
            int r = idx >> 5, c4 = (idx & 31) * 4;
            int gk = kb + r, gn = bcol + c4;
            float4 v = *(const float4*)(B + (size_t)gk * NMAT + gn);
            if (transformB) {                          // 1.5*I - 0.5*B fused
                v.x = -0.5f * v.x + ((gk == gn + 0) ? 1.5f : 0.0f);
                v.y = -0.5f * v.y + ((gk == gn + 1) ? 1.5f : 0.0f);
                v.z = -0.5f * v.z + ((gk == gn + 2) ? 1.5f : 0.0f);
                v.w = -0.5f * v.w + ((gk == gn + 3) ? 1.5f : 0.0f);
            }
            bR[i] = v;
        }
    };
    // ---- registers -> LDS buffer ------------------------------------------
    auto storeStage = [&](int buf) {
#pragma unroll
        for (int i = 0; i < 4; ++i) {
            int idx = tid + i * 128;
            int r = idx >> 3, c4 = (idx & 7) * 4;
            *(float4*)(&As[buf][r][c4]) = aR[i];
        }
#pragma unroll
        for (int i = 0; i < 8; ++i) {
            int idx = tid + i * 128;
            int r = idx >> 5, c4 = (idx & 31) * 4;
            *(float4*)(&Bs[buf][r][c4]) = bR[i];
        }
    };

    loadStage(0);
    storeStage(0);
    __syncthreads();

    const int NSTAGE = NMAT / BK;      // 32
    for (int s = 0; s < NSTAGE; ++s) {
        const int cur = s & 1;
        if (s + 1 < NSTAGE) loadStage((s + 1) * BK);  // prefetch next stage

        // ---- 8 x (K=4) WMMA steps from the LDS buffer selected by cur ---------------------------
#pragma unroll
        for (int kk = 0; kk < BK; kk += 4) {
            v2f af[4], bf[2];
#pragma unroll
            for (int i = 0; i < 4; ++i) {
                int m = i * 16 + l16;                     // A: row = lane&15
                af[i] = (v2f){ As[cur][m][kk + 2 * half], // lanes 0-15: K=0,1
                               As[cur][m][kk + 2 * half + 1] };
            }
#pragma unroll
            for (int j = 0; j < 2; ++j) {
                int n = wc * 32 + j * 16 + l16;
                bf[j] = (v2f){ Bs[cur][kk + 2 * half][n],
                               Bs[cur][kk + 2 * half + 1][n] };
            }
#pragma unroll
            for (int i = 0; i < 4; ++i)
#pragma unroll
                for (int j = 0; j < 2; ++j)
                    acc[i][j] = __builtin_amdgcn_wmma_f32_16x16x4_f32(
                        false, af[i], false, bf[j],
                        (short)0, acc[i][j], false, false);
        }

        if (s + 1 < NSTAGE) storeStage(cur ^ 1);      // fill other buffer
        __syncthreads();
    }

    // ---- epilogue: C/D layout VGPR r -> row r + 8*half, col = lane&15 -----
#pragma unroll
    for (int i = 0; i < 4; ++i) {
#pragma unroll
        for (int j = 0; j < 2; ++j) {
            int gm0 = brow + i * 16 + half * 8;
            int gn  = bcol + wc * 32 + j * 16 + l16;
#pragma unroll
            for (int r = 0; r < 8; ++r) {
                size_t off = (size_t)(gm0 + r) * NMAT + gn;
                float val = acc[i][j][r];
                if (epimode == 1) val = 0.5f * (val + E[off]);
                D[off] = val;
            }
        }
    }
}

// ---------------------------------------------------------------------------
// Elementwise kernels
// ---------------------------------------------------------------------------
__global__ __launch_bounds__(256)
void init_kernel(const float* __restrict__ W, float* __restrict__ Ws,
                 float* __restrict__ Z, float* __restrict__ U,
                 float* __restrict__ Mpre)
{
    int idx = blockIdx.x * 256 + threadIdx.x;
    int i = idx >> 10, j = idx & 1023;
    float ws  = 0.5f * (W[idx] + W[(size_t)j * NMAT + i]);  // sym(W)
    float eye = (i == j) ? 1.0f : 0.0f;
    Ws[idx] = ws;
    Z[idx]  = eye;            // Z0 = I
    U[idx]  = 0.0f;           // U0 = 0
    Mpre[idx] = ws + eye;     // M for first prox: Z0 - U0 + Ws
}

// M = sym(Mpre); deterministic per-block partial sums of M^2 (for ||M||_F)
__global__ __launch_bounds__(256)
void sym_sumsq_kernel(const float* __restrict__ Mpre, float* __restrict__ M,
                      float* __restrict__ partial)
{
    __shared__ float sdata[256];
    int idx = blockIdx.x * 256 + threadIdx.x;
    int i = idx >> 10, j = idx & 1023;
    float m = 0.5f * (Mpre[idx] + Mpre[(size_t)j * NMAT + i]);
    M[idx] = m;
    sdata[threadIdx.x] = m * m;
    __syncthreads();
    for (int s = 128; s > 0; s >>= 1) {
        if (threadIdx.x < s) sdata[threadIdx.x] += sdata[threadIdx.x + s];
        __syncthreads();
    }
    if (threadIdx.x == 0) partial[blockIdx.x] = sdata[0];
}

__global__ __launch_bounds__(256)
void reduce_kernel(const float* __restrict__ partial, float* __restrict__ scale)
{
    __shared__ float sdata[256];
    float s = 0.0f;
    for (int i = threadIdx.x; i < 4096; i += 256) s += partial[i];
    sdata[threadIdx.x] = s;
    __syncthreads();
    for (int k = 128; k > 0; k >>= 1) {
        if (threadIdx.x < k) sdata[threadIdx.x] += sdata[threadIdx.x + k];
        __syncthreads();
    }
    if (threadIdx.x == 0) {
        float fro = sqrtf(sdata[0]);
        scale[0] = (fro > 0.0f) ? (1.0f / fro) : 0.0f;
    }
}

__global__ __launch_bounds__(256)
void scale_kernel(const float* __restrict__ M, const float* __restrict__ scale,
                  float* __restrict__ Y)
{
    int idx = blockIdx.x * 256 + threadIdx.x;
    Y[idx] = M[idx] * scale[0];   // Y0 = M / ||M||_F  (spectral-norm bound)
}

// Z = proj_c(X+U); U += X - Z; Mpre = Z - U + Ws (next prox input)
__global__ __launch_bounds__(256)
void update_kernel(const float* __restrict__ X, float* __restrict__ U,
                   const float* __restrict__ Ws, float* __restrict__ Z,
                   float* __restrict__ Mpre)
{
    int idx = blockIdx.x * 256 + threadIdx.x;
    int i = idx >> 10, j = idx & 1023;
    float x  = X[idx];
    float u  = U[idx];
    float zc = x + u;
    float z  = (i == j) ? 1.0f : fmaxf(zc, 0.0f);
    float un = u + x - z;
    Z[idx]    = z;
    U[idx]    = un;
    Mpre[idx] = z - un + Ws[idx];
}

// ---------------------------------------------------------------------------
extern "C" void kernel_launch(void* const* d_in, const int* in_sizes, int n_in,
                              void* d_out, int out_size, void* d_ws, size_t ws_size,
                              hipStream_t stream)
{
    (void)in_sizes; (void)n_in; (void)out_size; (void)ws_size;

    const float* W = (const float*)d_in[0];   // 1024x1024 fp32 (d_in[1] = N)
    float* Z = (float*)d_out;                 // Z lives directly in d_out

    const size_t MSZ = (size_t)NMAT * NMAT;
    float* ws      = (float*)d_ws;
    float* Ws      = ws + 0 * MSZ;
    float* U       = ws + 1 * MSZ;
    float* Mpre    = ws + 2 * MSZ;
    float* M       = ws + 3 * MSZ;
    float* Ya      = ws + 4 * MSZ;
    float* Yb      = ws + 5 * MSZ;
    float* T       = ws + 6 * MSZ;
    float* X       = ws + 7 * MSZ;
    float* partial = ws + 8 * MSZ;            // 4096 floats
    float* scale   = partial + 4096;          // 1 float

    const dim3 eg(MSZ / 256), eb(256);
    const dim3 gg(NMAT / BN, NMAT / BM), gb(128);

    init_kernel<<<eg, eb, 0, stream>>>(W, Ws, Z, U, Mpre);

    for (int it = 0; it < ADMM_ITERS; ++it) {
        sym_sumsq_kernel<<<eg, eb, 0, stream>>>(Mpre, M, partial);
        reduce_kernel<<<1, 256, 0, stream>>>(partial, scale);
        scale_kernel<<<eg, eb, 0, stream>>>(M, scale, Ya);

        // Newton-Schulz sign iterations: Y <- 1.5*Y - 0.5*Y*(Y*Y)
        float* Y = Ya; float* Yn = Yb;
        for (int t = 0; t < NS_ITERS; ++t) {
            gemm_wmma_f32<<<gg, gb, 0, stream>>>(Y, Y, T, nullptr, 0, 0);
            gemm_wmma_f32<<<gg, gb, 0, stream>>>(Y, T, Yn, nullptr, 1, 0);
            float* tmp = Y; Y = Yn; Yn = tmp;
        }
        // X = 0.5*(M + sign(M)*M), fused in GEMM epilogue
        gemm_wmma_f32<<<gg, gb, 0, stream>>>(Y, M, X, M, 0, 1);

        update_kernel<<<eg, eb, 0, stream>>>(X, U, Ws, Z, Mpre);
    }
}